// MoeFc_58162447122834
// MI455X (gfx1250) — compile-verified
//
#include <hip/hip_runtime.h>
#include <hip/hip_bf16.h>

// Problem constants (from reference): B=4,S=1024 -> 4096 tokens, D=O=512, E=8, K=2
#define N_TOK   4096
#define DIM     512
#define NEXP    8
#define TOPK    2
#define TILE_M  64
#define TILE_N  128
#define KC      16
#define A_STRIDE 20          // 20*l15 mod 64 covers all multiples of 4 -> conflict-free b64 frags
#define MT_PER_E 64          // ceil(4096/64)

// Workspace layout (bytes); requires ws_size >= ~33.8 MB
#define OFF_COUNTS 0u
#define OFF_LISTS  256u
#define OFF_SLOTW  131328u
#define OFF_HA     164352u
#define OFF_HB     16941568u

typedef __attribute__((ext_vector_type(2))) float v2f;
typedef __attribute__((ext_vector_type(8))) float v8f;

__global__ __launch_bounds__(32) void moe_zero_kernel(int* __restrict__ counts) {
    if (threadIdx.x < NEXP) counts[threadIdx.x] = 0;
}

// One wave32 per token: logits -> softmax -> top-2 -> routing lists.
__global__ __launch_bounds__(128) void moe_gate_kernel(
    const float* __restrict__ x, const float* __restrict__ gw,
    const float* __restrict__ gb, int* __restrict__ counts,
    int* __restrict__ lists, float* __restrict__ slotw)
{
    const int wave = threadIdx.x >> 5;
    const int lane = threadIdx.x & 31;
    const int t = blockIdx.x * 4 + wave;
    if (t >= N_TOK) return;

    float acc[NEXP];
#pragma unroll
    for (int e = 0; e < NEXP; ++e) acc[e] = 0.f;

    const float* xr = x + (size_t)t * DIM;
    for (int c = lane; c < DIM; c += 32) {
        const float xv = xr[c];
        const float* g = gw + (size_t)c * NEXP;   // gate_w is [D,E] row-major
#pragma unroll
        for (int e = 0; e < NEXP; ++e) acc[e] = fmaf(xv, g[e], acc[e]);
    }
#pragma unroll
    for (int e = 0; e < NEXP; ++e) {
#pragma unroll
        for (int off = 16; off > 0; off >>= 1)
            acc[e] += __shfl_xor(acc[e], off, 32);
    }

    if (lane == 0) {
        float lg[NEXP], p[NEXP];
        float mx = -3.4e38f;
#pragma unroll
        for (int e = 0; e < NEXP; ++e) { lg[e] = acc[e] + gb[e]; mx = fmaxf(mx, lg[e]); }
        float s = 0.f;
#pragma unroll
        for (int e = 0; e < NEXP; ++e) { p[e] = __expf(lg[e] - mx); s += p[e]; }
        const float inv = 1.f / s;

        // top-2 (descending, ties -> lower index, matching jax.lax.top_k)
        int i0 = 0;
#pragma unroll
        for (int e = 1; e < NEXP; ++e) if (p[e] > p[i0]) i0 = e;
        int i1 = (i0 == 0) ? 1 : 0;
#pragma unroll
        for (int e = 0; e < NEXP; ++e) if (e != i0 && p[e] > p[i1]) i1 = e;

        // Faithful quirk: combine weights are probs[:, :K] (slot-indexed columns)
        slotw[t * 2 + 0] = p[0] * inv;
        slotw[t * 2 + 1] = p[1] * inv;

        int pos0 = atomicAdd(&counts[i0], 1);
        lists[i0 * N_TOK + pos0] = t * TOPK + 0;
        int pos1 = atomicAdd(&counts[i1], 1);
        lists[i1 * N_TOK + pos1] = t * TOPK + 1;
    }
}

// Swizzled B-tile index (floats): element (k, col) of the 16x128 chunk.
// Quad-interleave so each lane's (k,k+1) fragment pair is a contiguous,
// 8B-aligned b64; XOR of bits[3:2] by (col>>4)&3 spreads LDS banks.
__device__ __forceinline__ int boff(int k, int col) {
    return ((k >> 2) << 9) + ((((col << 2) + (k & 3))) ^ (((col >> 4) & 3) << 2));
}

// Grouped GEMM: rows gathered via per-expert lists, C = relu(A*W[e] + b[e]),
// scattered to dst by assignment id. K = N = 512 for all three layers.
// mode 0: A row for assignment a is x[a>>1]; mode 1: A row is Abase[a].
// Block = 128 threads (4 waves). Block tile 64x128; wave tile 16x128
// (8 v8f accumulators). Double-buffered global->reg->LDS staging.
__global__ __launch_bounds__(128) void moe_gemm_kernel(
    const float* __restrict__ Abase, int mode,
    const float* __restrict__ W, const float* __restrict__ bias,
    float* __restrict__ dst, const int* __restrict__ lists,
    const int* __restrict__ counts)
{
    const int e  = blockIdx.x / MT_PER_E;
    const int mt = blockIdx.x % MT_PER_E;
    const int cnt = counts[e];
    const int m0 = mt * TILE_M;
    if (m0 >= cnt) return;                       // uniform per block
    const int n0 = blockIdx.y * TILE_N;

    __shared__ float Al[TILE_M * A_STRIDE];      // 64 x 16 (padded to 20)
    __shared__ float Bl[KC * TILE_N];            // 16 x 128, swizzled (8 KB)

    const int tid  = threadIdx.x;
    const int lane = tid & 31;
    const int wv   = tid >> 5;
    const int half = lane >> 4;                  // 0: k,k+1   1: k+2,k+3
    const int l15  = lane & 15;
    const int* mylist = lists + e * N_TOK;
    const float* We   = W + (size_t)e * DIM * DIM;

    // ---- A staging plan: 64x16 floats = 128 thr x 2 float4 ----
    const float* asrc[2];
    int ast[2];
#pragma unroll
    for (int j = 0; j < 2; ++j) {
        const int i = tid + j * 128;
        const int r = i >> 2;
        const int aq = (i & 3) * 4;
        ast[j] = r * A_STRIDE + aq;              // 16B-aligned LDS float4 slot
        int m = m0 + r; if (m > cnt - 1) m = cnt - 1;
        const int a = mylist[m];
        const int row = (mode == 0) ? (a >> 1) : a;
        asrc[j] = Abase + (size_t)row * DIM + aq;
    }
    // ---- B staging plan: 16x128 floats = 128 thr x 4 float4 ----
    const float* bsrc[4];
    int bst[4];
#pragma unroll
    for (int j = 0; j < 4; ++j) {
        const int i = tid + j * 128;
        const int kr = i >> 5;                   // 0..15
        const int q  = (i & 31) * 4;             // 0..124
        bsrc[j] = We + (size_t)kr * DIM + n0 + q;
        bst[j]  = boff(kr, q);                   // cols q..q+3 are bst ^ (u<<2)
    }

    // ---- fragment read plan (loop-invariant LDS offsets) ----
    const int arowo = (wv * 16 + l15) * A_STRIDE + half * 2;
    int fco[8];
#pragma unroll
    for (int c = 0; c < 8; ++c) {
        const int col = c * 16 + l15;
        fco[c] = ((col << 2) + half * 2) ^ ((c & 3) << 2);
    }

    v8f acc[8] = {};

    // prefetch chunk 0 into registers
    float4 aPre[2], bPre[4];
#pragma unroll
    for (int j = 0; j < 2; ++j) aPre[j] = *(const float4*)(asrc[j]);
#pragma unroll
    for (int j = 0; j < 4; ++j) bPre[j] = *(const float4*)(bsrc[j]);

    for (int k0 = 0; k0 < DIM; k0 += KC) {
        __syncthreads();                         // LDS consumers done
#pragma unroll
        for (int j = 0; j < 2; ++j)
            *(float4*)&Al[ast[j]] = aPre[j];
#pragma unroll
        for (int j = 0; j < 4; ++j) {
            Bl[bst[j] ^ 0]  = bPre[j].x;
            Bl[bst[j] ^ 4]  = bPre[j].y;
            Bl[bst[j] ^ 8]  = bPre[j].z;
            Bl[bst[j] ^ 12] = bPre[j].w;
        }
        __syncthreads();                         // tiles visible

        if (k0 + KC < DIM) {                     // prefetch next chunk under compute
            const int kn = k0 + KC;
#pragma unroll
            for (int j = 0; j < 2; ++j) aPre[j] = *(const float4*)(asrc[j] + kn);
#pragma unroll
            for (int j = 0; j < 4; ++j) bPre[j] = *(const float4*)(bsrc[j] + (size_t)kn * DIM);
            if (k0 + 2 * KC < DIM)               // warm L2/WGP$ for chunk after next
                __builtin_prefetch(bsrc[0] + (size_t)(k0 + 2 * KC) * DIM, 0, 1);
        }

#pragma unroll
        for (int kk = 0; kk < KC; kk += 4) {
            const v2f af = *(const v2f*)&Al[arowo + kk];
#pragma unroll
            for (int c = 0; c < 8; ++c) {
                const v2f bf = *(const v2f*)&Bl[((kk >> 2) << 9) + fco[c]];
                acc[c] = __builtin_amdgcn_wmma_f32_16x16x4_f32(
                    false, af, false, bf, (short)0, acc[c], false, false);
            }
        }
    }

    // Epilogue: bias + relu, scatter rows by assignment id.
    const float* be = bias + e * DIM;
    int aidx[8]; bool avalid[8];
#pragma unroll
    for (int v = 0; v < 8; ++v) {
        const int r = m0 + wv * 16 + v + half * 8;
        avalid[v] = (r < cnt);
        aidx[v] = mylist[avalid[v] ? r : 0];
    }
#pragma unroll
    for (int c = 0; c < 8; ++c) {
        const int col = n0 + c * 16 + l15;
        const float bv = be[col];
#pragma unroll
        for (int v = 0; v < 8; ++v) {
            if (avalid[v])
                dst[(size_t)aidx[v] * DIM + col] = fmaxf(acc[c][v] + bv, 0.f);
        }
    }
}

// out[t,:] = w0[t]*y[2t,:] + w1[t]*y[2t+1,:]   (float4-vectorized)
__global__ __launch_bounds__(256) void moe_combine_kernel(
    const float* __restrict__ y, const float* __restrict__ slotw,
    float* __restrict__ out)
{
    const int i = blockIdx.x * 256 + threadIdx.x;   // one float4 per thread
    const int t = i >> 7;
    const int c = (i & 127) * 4;
    const float w0 = slotw[2 * t];
    const float w1 = slotw[2 * t + 1];
    const float4 y0 = *(const float4*)(y + (size_t)(2 * t) * DIM + c);
    const float4 y1 = *(const float4*)(y + (size_t)(2 * t + 1) * DIM + c);
    float4 o;
    o.x = w0 * y0.x + w1 * y1.x;
    o.y = w0 * y0.y + w1 * y1.y;
    o.z = w0 * y0.z + w1 * y1.z;
    o.w = w0 * y0.w + w1 * y1.w;
    *(float4*)(out + (size_t)t * DIM + c) = o;
}

extern "C" void kernel_launch(void* const* d_in, const int* in_sizes, int n_in,
                              void* d_out, int out_size, void* d_ws, size_t ws_size,
                              hipStream_t stream) {
    const float* x  = (const float*)d_in[0];
    const float* gw = (const float*)d_in[1];
    const float* gb = (const float*)d_in[2];
    const float* w1 = (const float*)d_in[3];
    const float* b1 = (const float*)d_in[4];
    const float* w2 = (const float*)d_in[5];
    const float* b2 = (const float*)d_in[6];
    const float* w3 = (const float*)d_in[7];
    const float* b3 = (const float*)d_in[8];

    char* ws = (char*)d_ws;
    int*   counts = (int*)(ws + OFF_COUNTS);
    int*   lists  = (int*)(ws + OFF_LISTS);
    float* slotw  = (float*)(ws + OFF_SLOTW);
    float* hA     = (float*)(ws + OFF_HA);
    float* hB     = (float*)(ws + OFF_HB);
    float* out    = (float*)d_out;

    moe_zero_kernel<<<1, 32, 0, stream>>>(counts);
    moe_gate_kernel<<<N_TOK / 4, 128, 0, stream>>>(x, gw, gb, counts, lists, slotw);

    dim3 g(NEXP * MT_PER_E, DIM / TILE_N);
    moe_gemm_kernel<<<g, 128, 0, stream>>>(x,  0, w1, b1, hA, lists, counts);
    moe_gemm_kernel<<<g, 128, 0, stream>>>(hA, 1, w2, b2, hB, lists, counts);
    moe_gemm_kernel<<<g, 128, 0, stream>>>(hB, 1, w3, b3, hA, lists, counts);

    moe_combine_kernel<<<(N_TOK * DIM) / 1024, 256, 0, stream>>>(hA, slotw, out);
}